// EGCN_33758442947102
// MI455X (gfx1250) — compile-verified
//
#include <hip/hip_runtime.h>
#include <hip/hip_bf16.h>
#include <math.h>

#define NODES 100000
#define EDGES 1600000
#define IN_F  128
#define EF_F  32
#define H1_F  64
#define OUT_F 64
#define SLOPE 0.2f

typedef __attribute__((ext_vector_type(2))) float v2f;
typedef __attribute__((ext_vector_type(8))) float v8f;

#define WMMA_F32_16x16x4(A, B, C) \
  __builtin_amdgcn_wmma_f32_16x16x4_f32(false, (A), false, (B), (short)0, (C), false, false)

__device__ __forceinline__ void atomicMaxFloat(float* addr, float val) {
  // Standard sign-split trick: works for all orderings of float via int/uint atomics.
  if (val >= 0.0f) atomicMax((int*)addr, __float_as_int(val));
  else             atomicMin((unsigned int*)addr, __float_as_uint(val));
}

// ---------------------------------------------------------------------------
// wea[j] = sum_h We[j][h] * ae[h]   (collapses (ef@We)@ae to ef@wea)
// ---------------------------------------------------------------------------
__global__ void compute_wea(const float* __restrict__ We, const float* __restrict__ ae,
                            float* __restrict__ wea) {
  int j = threadIdx.x;
  if (j < EF_F) {
    float s = 0.f;
    #pragma unroll
    for (int t = 0; t < H1_F; ++t) s += We[j * H1_F + t] * ae[t];
    wea[j] = s;
  }
}

// ---------------------------------------------------------------------------
// Layer-1 fused GEMM: h = feat @ Wf ; rst = feat @ Wres + b1
// One wave -> 16x64 output strip, f32 WMMA (16x16x4), K = 128.
// A 16x4 frag: lane l holds A[l%16][kb], A[l%16][kb+1], kb = (l/16)*2.
// B 4x16 frag: lane l holds B[kb][l%16], B[kb+1][l%16] (symmetric layout).
// C/D 16x16: vgpr v -> row v (lanes 0-15) / row v+8 (lanes 16-31), col = l%16.
// ---------------------------------------------------------------------------
__global__ __launch_bounds__(128) void gemm_layer1(
    const float* __restrict__ feat, const float* __restrict__ Wf,
    const float* __restrict__ Wres, const float* __restrict__ b1,
    float* __restrict__ h, float* __restrict__ rst) {
  const int lane = threadIdx.x & 31;
  const int wave = threadIdx.x >> 5;
  const int rowTile = blockIdx.x * 4 + wave;
  if (rowTile >= NODES / 16) return;
  const int row0 = rowTile * 16;
  const int am = lane & 15;
  const int kb = (lane >> 4) << 1;   // 0 or 2
  const int bn = lane & 15;

  v8f aH0 = {}, aH1 = {}, aH2 = {}, aH3 = {};
  v8f aR0 = {}, aR1 = {}, aR2 = {}, aR3 = {};
  const float* arow = feat + (size_t)(row0 + am) * IN_F;

  #pragma unroll 4
  for (int k = 0; k < IN_F; k += 4) {
    v2f a; a.x = arow[k + kb]; a.y = arow[k + kb + 1];
    const float* bf0 = Wf   + (size_t)(k + kb) * H1_F + bn;
    const float* bf1 = Wf   + (size_t)(k + kb + 1) * H1_F + bn;
    const float* br0 = Wres + (size_t)(k + kb) * H1_F + bn;
    const float* br1 = Wres + (size_t)(k + kb + 1) * H1_F + bn;
    v2f b;
    b.x = bf0[0];  b.y = bf1[0];  aH0 = WMMA_F32_16x16x4(a, b, aH0);
    b.x = bf0[16]; b.y = bf1[16]; aH1 = WMMA_F32_16x16x4(a, b, aH1);
    b.x = bf0[32]; b.y = bf1[32]; aH2 = WMMA_F32_16x16x4(a, b, aH2);
    b.x = bf0[48]; b.y = bf1[48]; aH3 = WMMA_F32_16x16x4(a, b, aH3);
    b.x = br0[0];  b.y = br1[0];  aR0 = WMMA_F32_16x16x4(a, b, aR0);
    b.x = br0[16]; b.y = br1[16]; aR1 = WMMA_F32_16x16x4(a, b, aR1);
    b.x = br0[32]; b.y = br1[32]; aR2 = WMMA_F32_16x16x4(a, b, aR2);
    b.x = br0[48]; b.y = br1[48]; aR3 = WMMA_F32_16x16x4(a, b, aR3);
  }

  const int mb = (lane >> 4) * 8;
  const float c0 = b1[bn], c1 = b1[16 + bn], c2 = b1[32 + bn], c3 = b1[48 + bn];
  #pragma unroll
  for (int v = 0; v < 8; ++v) {
    size_t r = (size_t)(row0 + mb + v) * H1_F;
    h[r + bn]        = aH0[v];
    h[r + 16 + bn]   = aH1[v];
    h[r + 32 + bn]   = aH2[v];
    h[r + 48 + bn]   = aH3[v];
    rst[r + bn]      = aR0[v] + c0;
    rst[r + 16 + bn] = aR1[v] + c1;
    rst[r + 32 + bn] = aR2[v] + c2;
    rst[r + 48 + bn] = aR3[v] + c3;
  }
}

// ---------------------------------------------------------------------------
// Layer-2 GEMM with ReLU on load: h2 = relu(rst) @ Wg, K = 64.
// ---------------------------------------------------------------------------
__global__ __launch_bounds__(128) void gemm_layer2(
    const float* __restrict__ rst, const float* __restrict__ Wg,
    float* __restrict__ h2) {
  const int lane = threadIdx.x & 31;
  const int wave = threadIdx.x >> 5;
  const int rowTile = blockIdx.x * 4 + wave;
  if (rowTile >= NODES / 16) return;
  const int row0 = rowTile * 16;
  const int am = lane & 15;
  const int kb = (lane >> 4) << 1;
  const int bn = lane & 15;

  v8f a0 = {}, a1 = {}, a2 = {}, a3 = {};
  const float* arow = rst + (size_t)(row0 + am) * H1_F;

  #pragma unroll 4
  for (int k = 0; k < H1_F; k += 4) {
    v2f a;
    a.x = fmaxf(arow[k + kb], 0.f);
    a.y = fmaxf(arow[k + kb + 1], 0.f);
    const float* b0 = Wg + (size_t)(k + kb) * OUT_F + bn;
    const float* b1p = Wg + (size_t)(k + kb + 1) * OUT_F + bn;
    v2f b;
    b.x = b0[0];  b.y = b1p[0];  a0 = WMMA_F32_16x16x4(a, b, a0);
    b.x = b0[16]; b.y = b1p[16]; a1 = WMMA_F32_16x16x4(a, b, a1);
    b.x = b0[32]; b.y = b1p[32]; a2 = WMMA_F32_16x16x4(a, b, a2);
    b.x = b0[48]; b.y = b1p[48]; a3 = WMMA_F32_16x16x4(a, b, a3);
  }

  const int mb = (lane >> 4) * 8;
  #pragma unroll
  for (int v = 0; v < 8; ++v) {
    size_t r = (size_t)(row0 + mb + v) * OUT_F;
    h2[r + bn]      = a0[v];
    h2[r + 16 + bn] = a1[v];
    h2[r + 32 + bn] = a2[v];
    h2[r + 48 + bn] = a3[v];
  }
}

// ---------------------------------------------------------------------------
// Per-node attention projections: el = h.al, er = h.ar
// ---------------------------------------------------------------------------
__global__ void attn_scores(const float* __restrict__ h, const float* __restrict__ al,
                            const float* __restrict__ ar, float* __restrict__ el,
                            float* __restrict__ er) {
  int i = blockIdx.x * blockDim.x + threadIdx.x;
  if (i >= NODES) return;
  const float4* hr = (const float4*)(h + (size_t)i * H1_F);
  const float4* a4 = (const float4*)al;
  const float4* r4 = (const float4*)ar;
  float sl = 0.f, sr = 0.f;
  #pragma unroll
  for (int j = 0; j < H1_F / 4; ++j) {
    float4 hv = hr[j], av = a4[j], rv = r4[j];
    sl += hv.x * av.x + hv.y * av.y + hv.z * av.z + hv.w * av.w;
    sr += hv.x * rv.x + hv.y * rv.y + hv.z * rv.z + hv.w * rv.w;
  }
  el[i] = sl; er[i] = sr;
}

__global__ void init_softmax(float* __restrict__ m, float* __restrict__ s) {
  int i = blockIdx.x * blockDim.x + threadIdx.x;
  if (i >= NODES) return;
  m[i] = __int_as_float(0xff800000);  // -inf
  s[i] = 0.f;
}

__global__ void zero_buf(float* __restrict__ p, int n) {
  int i = blockIdx.x * blockDim.x + threadIdx.x;
  if (i < n) p[i] = 0.f;
}

// ---------------------------------------------------------------------------
// Edge pass A: raw score + atomic segment-max
// ---------------------------------------------------------------------------
template <bool WITH_EDGE>
__global__ void edge_scores(const int* __restrict__ src, const int* __restrict__ dst,
                            const float* __restrict__ edge_feat, const float* __restrict__ wea,
                            const float* __restrict__ el, const float* __restrict__ er,
                            float* __restrict__ ew, float* __restrict__ m) {
  int i = blockIdx.x * blockDim.x + threadIdx.x;
  if (i >= EDGES) return;
  int si = src[i], di = dst[i];
  float e = el[si] + er[di];
  if (WITH_EDGE) {
    const float4* ef = (const float4*)(edge_feat + (size_t)i * EF_F);
    const float4* w4 = (const float4*)wea;
    float s = 0.f;
    #pragma unroll
    for (int j = 0; j < EF_F / 4; ++j) {
      float4 x = ef[j], w = w4[j];
      s += x.x * w.x + x.y * w.y + x.z * w.z + x.w * w.w;
    }
    e += s;
  }
  e = (e > 0.f) ? e : SLOPE * e;   // leaky_relu
  ew[i] = e;
  atomicMaxFloat(&m[di], e);
}

// ---------------------------------------------------------------------------
// Edge pass B: ex = exp(e - m[dst]); segment-sum via atomics
// ---------------------------------------------------------------------------
__global__ void edge_exp(const int* __restrict__ dst, float* __restrict__ ew,
                         const float* __restrict__ m, float* __restrict__ s) {
  int i = blockIdx.x * blockDim.x + threadIdx.x;
  if (i >= EDGES) return;
  int di = dst[i];
  float ex = __expf(ew[i] - m[di]);
  ew[i] = ex;
  atomicAdd(&s[di], ex);
}

// ---------------------------------------------------------------------------
// Edge pass C: acc[dst,:] += (ex/s[dst]) * h[src,:]  (16 threads/edge, float4)
// ---------------------------------------------------------------------------
__global__ void edge_aggregate(const int* __restrict__ src, const int* __restrict__ dst,
                               const float* __restrict__ ew, const float* __restrict__ s,
                               const float* __restrict__ h, float* __restrict__ acc) {
  int t = blockIdx.x * blockDim.x + threadIdx.x;
  int e = t >> 4;
  if (e >= EDGES) return;
  int f = t & 15;
  int si = src[e], di = dst[e];
  float alpha = ew[e] / s[di];
  float4 hv = ((const float4*)(h + (size_t)si * H1_F))[f];
  float* o = acc + (size_t)di * H1_F + f * 4;
  atomicAdd(o + 0, alpha * hv.x);
  atomicAdd(o + 1, alpha * hv.y);
  atomicAdd(o + 2, alpha * hv.z);
  atomicAdd(o + 3, alpha * hv.w);
}

__global__ void add_bias(float* __restrict__ out, const float* __restrict__ b) {
  int i = blockIdx.x * blockDim.x + threadIdx.x;
  if (i < NODES * OUT_F) out[i] += b[i & (OUT_F - 1)];
}

// ---------------------------------------------------------------------------
extern "C" void kernel_launch(void* const* d_in, const int* in_sizes, int n_in,
                              void* d_out, int out_size, void* d_ws, size_t ws_size,
                              hipStream_t stream) {
  const float* feat      = (const float*)d_in[0];
  const float* edge_feat = (const float*)d_in[1];
  const int*   src       = (const int*)d_in[2];
  const int*   dst       = (const int*)d_in[3];
  const float* Wf        = (const float*)d_in[4];
  const float* We        = (const float*)d_in[5];
  const float* al        = (const float*)d_in[6];
  const float* ar        = (const float*)d_in[7];
  const float* ae        = (const float*)d_in[8];
  const float* Wres      = (const float*)d_in[9];
  const float* b1        = (const float*)d_in[10];
  const float* Wg        = (const float*)d_in[11];
  const float* al2       = (const float*)d_in[12];
  const float* ar2       = (const float*)d_in[13];
  const float* b2        = (const float*)d_in[14];
  float* out = (float*)d_out;

  float* ws  = (float*)d_ws;
  float* h   = ws;                         // N*64 (reused as h2 for layer 2)
  float* rst = h   + (size_t)NODES * 64;   // N*64
  float* ew  = rst + (size_t)NODES * 64;   // E
  float* el  = ew  + EDGES;                // N
  float* er  = el  + NODES;                // N
  float* mb  = er  + NODES;                // N
  float* sb  = mb  + NODES;                // N
  float* wea = sb  + NODES;                // 32

  const int gn  = (NODES + 255) / 256;
  const int ge  = (EDGES + 255) / 256;
  const int ga  = (EDGES * 16 + 255) / 256;
  const int gg  = (NODES / 16 + 3) / 4;    // 6250 row tiles, 4 waves/block
  const int gnf = (NODES * 64 + 255) / 256;

  // ---- Layer 1: EdgeGATConv ----
  compute_wea<<<1, 32, 0, stream>>>(We, ae, wea);
  gemm_layer1<<<gg, 128, 0, stream>>>(feat, Wf, Wres, b1, h, rst);
  attn_scores<<<gn, 256, 0, stream>>>(h, al, ar, el, er);
  init_softmax<<<gn, 256, 0, stream>>>(mb, sb);
  edge_scores<true><<<ge, 256, 0, stream>>>(src, dst, edge_feat, wea, el, er, ew, mb);
  edge_exp<<<ge, 256, 0, stream>>>(dst, ew, mb, sb);
  edge_aggregate<<<ga, 256, 0, stream>>>(src, dst, ew, sb, h, rst); // rst already = feat@Wres + b1

  // ---- Layer 2: GATConv (ReLU fused into GEMM load) ----
  gemm_layer2<<<gg, 128, 0, stream>>>(rst, Wg, h);                  // h now holds h2
  attn_scores<<<gn, 256, 0, stream>>>(h, al2, ar2, el, er);
  init_softmax<<<gn, 256, 0, stream>>>(mb, sb);
  zero_buf<<<gnf, 256, 0, stream>>>(out, NODES * 64);
  edge_scores<false><<<ge, 256, 0, stream>>>(src, dst, nullptr, nullptr, el, er, ew, mb);
  edge_exp<<<ge, 256, 0, stream>>>(dst, ew, mb, sb);
  edge_aggregate<<<ga, 256, 0, stream>>>(src, dst, ew, sb, h, out);
  add_bias<<<gnf, 256, 0, stream>>>(out, b2);
}